// PtModule_73667279061093
// MI455X (gfx1250) — compile-verified
//
#include <hip/hip_runtime.h>

// Elementwise: y = (x + 1) * 2 / 3 ; out = (y > 0) ? y - 5 : y
// 8192x8192 f32: pure HBM-streaming (AI ~ 0.5 flop/byte, floor ~22us @ 23.3TB/s).
// Strategy: wave32 blocks; each thread owns a contiguous 128B chunk so all
// 8 B128 NT loads/stores use immediate offsets off one base address (no
// per-access 64-bit address math), giving an 8-deep clause of outstanding
// loads per thread.

typedef __attribute__((ext_vector_type(4))) float v4f;

// (x+1)*2/3 with the division folded into one positive constant multiply.
// Safe: multiplying by a positive constant preserves sign exactly, so the
// (y > 0) select matches the two-step reference for every input; magnitudes
// differ by at most 1 ulp.
#define EW_SCALE (2.0f / 3.0f)

__device__ __forceinline__ float ew_op(float x) {
    float y = (x + 1.0f) * EW_SCALE;
    return (y > 0.0f) ? (y - 5.0f) : y;
}

__device__ __forceinline__ v4f ew_op4(v4f v) {
    v4f r;
    r.x = ew_op(v.x);
    r.y = ew_op(v.y);
    r.z = ew_op(v.z);
    r.w = ew_op(v.w);
    return r;
}

constexpr int V4_PER_THREAD = 8;   // 8 x 16B = 128B contiguous per thread

__global__ __launch_bounds__(256) void ew_stream_kernel(
    const float* __restrict__ x, float* __restrict__ out, long long n) {
    const long long n4   = n >> 2;
    const long long tid  = (long long)blockIdx.x * blockDim.x + threadIdx.x;
    const long long base = tid * V4_PER_THREAD;          // float4 index

    const v4f* __restrict__ xv = (const v4f*)x;
    v4f* __restrict__ ov       = (v4f*)out;

    if (base + V4_PER_THREAD <= n4) {
        // Fast path: 8 B128 NT loads at immediate offsets 0..112 from one
        // base address -> single s_clause, 8 loads in flight, then math,
        // then 8 B128 NT stores.
        v4f r[V4_PER_THREAD];
#pragma unroll
        for (int k = 0; k < V4_PER_THREAD; ++k)
            r[k] = __builtin_nontemporal_load(xv + base + k);
#pragma unroll
        for (int k = 0; k < V4_PER_THREAD; ++k)
            r[k] = ew_op4(r[k]);
#pragma unroll
        for (int k = 0; k < V4_PER_THREAD; ++k)
            __builtin_nontemporal_store(r[k], ov + base + k);
    } else {
        // Tail: this thread's chunk overlaps the end of the array (or n%4
        // leftovers). Scalar within this thread's own range only.
        const long long lo = base << 2;
        long long hi = (base + V4_PER_THREAD) << 2;
        if (hi > n) hi = n;
        for (long long j = lo; j < hi; ++j)
            out[j] = ew_op(x[j]);
    }
}

extern "C" void kernel_launch(void* const* d_in, const int* in_sizes, int n_in,
                              void* d_out, int out_size, void* d_ws, size_t ws_size,
                              hipStream_t stream) {
    (void)n_in; (void)d_ws; (void)ws_size; (void)out_size;
    const float* x = (const float*)d_in[0];
    float* out     = (float*)d_out;
    const long long n = (long long)in_sizes[0];   // 8192*8192 = 64Mi

    // Each block covers 256 threads * 128B = 32KB of the tensor.
    const int  threads          = 256;
    const long long per_block   = (long long)threads * V4_PER_THREAD * 4; // elements
    const int  blocks           = (int)((n + per_block - 1) / per_block); // 8192
    ew_stream_kernel<<<blocks, threads, 0, stream>>>(x, out, n);
}